// BeamCharacterDecoder_35880156790962
// MI455X (gfx1250) — compile-verified
//
#include <hip/hip_runtime.h>
#include <stdint.h>

// ---- problem constants (from reference) ----
#define TT 128
#define BB 8
#define CC 25000
#define WW 4
#define EOS_IDX 1

// phase-1 streaming: 256 threads * 4 floats = 1024 floats per block-iteration
#define CHUNK 1024
#define NITER ((CC + CHUNK - 1) / CHUNK)   // 25
#define NBUF 4                              // async pipeline depth

typedef __attribute__((address_space(3))) void lds_void_t;

__device__ __forceinline__ uint32_t lds_off_u32(void* p) {
  // generic -> LDS addrspace cast; AS3 pointers are 32-bit LDS byte offsets
  return (uint32_t)(uintptr_t)(lds_void_t*)p;
}

// CDNA5 async global->LDS copy (tracked by ASYNCcnt). GV addressing mode.
// NT hint: logits are read exactly once; don't churn L2 with 102 MB.
__device__ __forceinline__ void async_copy_b128(void* lds_dst, const void* gsrc) {
  asm volatile("global_load_async_to_lds_b128 %0, %1, off th:TH_LOAD_NT"
               :: "v"(lds_off_u32(lds_dst)), "v"((uint64_t)(uintptr_t)gsrc)
               : "memory");
}
// wait until at most n async ops outstanding (n known per call site, 0..3)
__device__ __forceinline__ void wait_async_le(int n) {
  switch (n) {
    case 0: asm volatile("s_wait_asynccnt 0x0" ::: "memory"); break;
    case 1: asm volatile("s_wait_asynccnt 0x1" ::: "memory"); break;
    case 2: asm volatile("s_wait_asynccnt 0x2" ::: "memory"); break;
    default: asm volatile("s_wait_asynccnt 0x3" ::: "memory"); break;
  }
}

// top-k comparator matching jax.lax.top_k tie-breaking (value desc, index asc)
__device__ __forceinline__ bool kbetter(float v, int i, float v2, int i2) {
  return (v > v2) || (v == v2 && i < i2);
}
__device__ __forceinline__ void kinsert(float (&tv)[4], int (&ti)[4], float v, int c) {
  if (!kbetter(v, c, tv[3], ti[3])) return;           // fast reject (common case)
  if (kbetter(v, c, tv[0], ti[0])) {
    tv[3]=tv[2]; ti[3]=ti[2]; tv[2]=tv[1]; ti[2]=ti[1];
    tv[1]=tv[0]; ti[1]=ti[0]; tv[0]=v; ti[0]=c;
  } else if (kbetter(v, c, tv[1], ti[1])) {
    tv[3]=tv[2]; ti[3]=ti[2]; tv[2]=tv[1]; ti[2]=ti[1]; tv[1]=v; ti[1]=c;
  } else if (kbetter(v, c, tv[2], ti[2])) {
    tv[3]=tv[2]; ti[3]=ti[2]; tv[2]=v; ti[2]=c;
  } else {
    tv[3]=v; ti[3]=c;
  }
}

// -------------------------------------------------------------------------
// Phase 1: for each (t,b) row of 25000 logits, compute online softmax (m,s)
// and top-4 (logit, index); emit per-row {p[4], logp[4], char[4]}.
// One block per row; 4-deep async global->LDS pipeline (3 chunks in flight
// past the one being consumed); each wave consumes only the LDS slice it
// loaded itself, so the streaming loop needs no barriers.
// -------------------------------------------------------------------------
__global__ __launch_bounds__(256) void tb_topk_softmax(
    const float* __restrict__ logits,
    float* __restrict__ wsP, float* __restrict__ wsLP, int* __restrict__ wsC) {
  __shared__ __align__(16) float sbuf[NBUF][CHUNK];
  __shared__ float red_m[8], red_s[8];
  __shared__ float red_tv[8][4];
  __shared__ int   red_ti[8][4];

  const int tb = blockIdx.x;                      // t*B + b
  const float* __restrict__ src = logits + (size_t)tb * CC;
  const int lane = threadIdx.x & 31;              // wave32
  const int wave = threadIdx.x >> 5;              // 8 waves
  const int slot = wave * 128 + lane * 4;         // float offset within chunk

  float m = -3.0e38f, s = 0.0f;
  float tv[4]; int ti[4];
#pragma unroll
  for (int k = 0; k < 4; ++k) { tv[k] = -3.0e38f; ti[k] = 0x7FFFFFFF; }

  // prime the pipeline with NBUF-1 chunks (all in-range: 3*CHUNK < CC)
#pragma unroll
  for (int pk = 0; pk < NBUF - 1; ++pk) {
    async_copy_b128(&sbuf[pk][slot], src + pk * CHUNK + slot);
  }

  for (int it = 0; it < NITER; ++it) {
    const int buf = it & (NBUF - 1);
    if (it + (NBUF - 1) < NITER) {
      int c0 = (it + (NBUF - 1)) * CHUNK + slot;
      if (c0 > CC - 4) c0 = CC - 4;               // tail: clamp (C % 4 == 0)
      async_copy_b128(&sbuf[(it + (NBUF - 1)) & (NBUF - 1)][slot], src + c0);
    }
    {
      const int rem = NITER - 1 - it;             // chunks issued beyond this one
      wait_async_le(rem < (NBUF - 1) ? rem : (NBUF - 1));
    }
    const int base = it * CHUNK + slot;
    float4 v4 = *reinterpret_cast<const float4*>(&sbuf[buf][slot]);
    float vals[4] = {v4.x, v4.y, v4.z, v4.w};
#pragma unroll
    for (int j = 0; j < 4; ++j) {
      const int c = base + j;
      const float v = (c < CC) ? vals[j] : -3.0e38f;   // masked tail
      if (v > m) { s = s * __expf(m - v) + 1.0f; m = v; }
      else       { s += __expf(v - m); }
      kinsert(tv, ti, v, c);
    }
  }

  // wave32 butterfly reduction of (m,s) and top-4 lists
#pragma unroll
  for (int off = 16; off >= 1; off >>= 1) {
    const float om = __shfl_xor(m, off, 32);
    const float os = __shfl_xor(s, off, 32);
    float ov[4]; int oi[4];
#pragma unroll
    for (int k = 0; k < 4; ++k) {
      ov[k] = __shfl_xor(tv[k], off, 32);
      oi[k] = __shfl_xor(ti[k], off, 32);
    }
    const float nm = fmaxf(m, om);
    s = s * __expf(m - nm) + os * __expf(om - nm);
    m = nm;
#pragma unroll
    for (int k = 0; k < 4; ++k) kinsert(tv, ti, ov[k], oi[k]);
  }

  if (lane == 0) {
    red_m[wave] = m; red_s[wave] = s;
#pragma unroll
    for (int k = 0; k < 4; ++k) { red_tv[wave][k] = tv[k]; red_ti[wave][k] = ti[k]; }
  }
  __syncthreads();

  if (threadIdx.x == 0) {
    float M = red_m[0], S = red_s[0];
    float Tv[4]; int Ti[4];
#pragma unroll
    for (int k = 0; k < 4; ++k) { Tv[k] = red_tv[0][k]; Ti[k] = red_ti[0][k]; }
    for (int w = 1; w < 8; ++w) {
      const float om = red_m[w], os = red_s[w];
      const float nm = fmaxf(M, om);
      S = S * __expf(M - nm) + os * __expf(om - nm);
      M = nm;
#pragma unroll
      for (int k = 0; k < 4; ++k) kinsert(Tv, Ti, red_tv[w][k], red_ti[w][k]);
    }
    const float logS = __logf(S);
    const int base = tb * 4;
#pragma unroll
    for (int k = 0; k < 4; ++k) {
      wsP[base + k]  = __expf(Tv[k] - M) / S;       // softmax prob of top-k char
      wsLP[base + k] = (Tv[k] - M) - logS;          // its log-prob
      wsC[base + k]  = Ti[k];                       // char index
    }
  }
}

// -------------------------------------------------------------------------
// Phase 2: exact beam recurrence over the per-step top-4 tables.
// 1 wave; lane b (<8) runs batch b serially over T. Sequences live in LDS
// (double-buffered for the backpointer gather). Faithful semantics:
//  - probs replicated across active beams (softmax is shift-invariant),
//    inactive beams masked to -1 (never selected since probs >= 0, C >= W)
//  - flattened top-k over W*C with (value desc, flat index asc) tie-break
//  - new_logs = logs[beam] + log(top_p); EOS stored as BLANK at slot t+1
//  - stable re-sort by cumulative log (desc) each step; final stable sort
//    by last-step prob score.
// -------------------------------------------------------------------------
__global__ void beam_combine(const float* __restrict__ wsP,
                             const float* __restrict__ wsLP,
                             const int* __restrict__ wsC,
                             float* __restrict__ out) {
  __shared__ int seq[2][BB][WW][TT + 1];
  const int lane = threadIdx.x;
  for (int i = lane; i < 2 * BB * WW * (TT + 1); i += 32)
    (&seq[0][0][0][0])[i] = 0;                       // BLANK init
  __syncthreads();
  if (lane >= BB) return;
  const int b = lane;

  float scores[WW] = {0.f, 0.f, 0.f, 0.f};
  float logs[WW]   = {0.f, 0.f, 0.f, 0.f};
  bool  active[WW] = {true, false, false, false};
  int cur = 0;

  for (int t = 0; t < TT; ++t) {
    const int base = (t * BB + b) * 4;
    float p[4], lp[4]; int ch[4];
#pragma unroll
    for (int k = 0; k < 4; ++k) {
      p[k] = wsP[base + k]; lp[k] = wsLP[base + k]; ch[k] = wsC[base + k];
    }
    // candidate set: per active beam, that beam's copy of the row-top-4
    float cv[16], clp[16]; int cw[16], cc[16], cf[16];
    int n = 0;
    for (int w = 0; w < WW; ++w) {
      if (!active[w]) continue;
      for (int k = 0; k < 4; ++k) {
        cv[n] = p[k]; clp[n] = lp[k]; cw[n] = w; cc[n] = ch[k];
        cf[n] = w * CC + ch[k];                      // flat index for tie-break
        ++n;
      }
    }
    // top-WW selection: value desc, flat index asc (lax.top_k tie rule)
    for (int sel = 0; sel < WW; ++sel) {
      int bi = sel;
      for (int j = sel + 1; j < n; ++j)
        if (cv[j] > cv[bi] || (cv[j] == cv[bi] && cf[j] < cf[bi])) bi = j;
      float tf; int tmp;
      tf = cv[sel];  cv[sel] = cv[bi];  cv[bi] = tf;
      tf = clp[sel]; clp[sel] = clp[bi]; clp[bi] = tf;
      tmp = cw[sel]; cw[sel] = cw[bi];  cw[bi] = tmp;
      tmp = cc[sel]; cc[sel] = cc[bi];  cc[bi] = tmp;
      tmp = cf[sel]; cf[sel] = cf[bi];  cf[bi] = tmp;
    }
    float nsc[WW], nlg[WW]; int bw[WW], nch[WW];
    for (int k = 0; k < WW; ++k) {
      bw[k]  = cw[k];
      nsc[k] = cv[k];
      nlg[k] = logs[bw[k]] + clp[k];
      nch[k] = (cc[k] == EOS_IDX) ? 0 : cc[k];       // eos -> blank
    }
    // stable argsort by nlg descending (== stable argsort of -nlg)
    int ord[WW] = {0, 1, 2, 3};
    for (int i = 1; i < WW; ++i) {
      const int key = ord[i]; int j = i - 1;
      while (j >= 0 && nlg[ord[j]] < nlg[key]) { ord[j + 1] = ord[j]; --j; }
      ord[j + 1] = key;
    }
    // gather (backpointer) + append + permute into the other buffer
    const int nxt = cur ^ 1;
    for (int k = 0; k < WW; ++k) {
      const int sc = ord[k];
      for (int i = 0; i <= TT; ++i) seq[nxt][b][k][i] = seq[cur][b][bw[sc]][i];
      seq[nxt][b][k][t + 1] = nch[sc];
      scores[k] = nsc[sc];
    }
    float nl[WW];
    for (int k = 0; k < WW; ++k) nl[k] = nlg[ord[k]];
    for (int k = 0; k < WW; ++k) logs[k] = nl[k];
    for (int w = 0; w < WW; ++w) active[w] = true;
    cur = nxt;
  }

  // final stable sort by last-step prob score (desc)
  int ord[WW] = {0, 1, 2, 3};
  for (int i = 1; i < WW; ++i) {
    const int key = ord[i]; int j = i - 1;
    while (j >= 0 && scores[ord[j]] < scores[key]) { ord[j + 1] = ord[j]; --j; }
    ord[j + 1] = key;
  }
  float* outSeq = out;                               // [B, W, T+1] as float
  float* outSc  = out + BB * WW * (TT + 1);          // [B, W]
  float* outLg  = outSc + BB * WW;                   // [B, W]
  for (int w = 0; w < WW; ++w) {
    const int sc = ord[w];
    for (int i = 0; i <= TT; ++i)
      outSeq[(b * WW + w) * (TT + 1) + i] = (float)seq[cur][b][sc][i];
    outSc[b * WW + w] = scores[sc];
    outLg[b * WW + w] = logs[sc];
  }
}

extern "C" void kernel_launch(void* const* d_in, const int* in_sizes, int n_in,
                              void* d_out, int out_size, void* d_ws, size_t ws_size,
                              hipStream_t stream) {
  (void)in_sizes; (void)n_in; (void)out_size; (void)ws_size;
  const float* logits = (const float*)d_in[0];       // [T, B, C] f32
  // d_in[1] = seq_len: all rows are full length T; unused by the math.

  float* wsP  = (float*)d_ws;                        // [T*B*4] probs
  float* wsLP = wsP  + (size_t)TT * BB * 4;          // [T*B*4] log-probs
  int*   wsC  = (int*)(wsLP + (size_t)TT * BB * 4);  // [T*B*4] char indices

  tb_topk_softmax<<<TT * BB, 256, 0, stream>>>(logits, wsP, wsLP, wsC);
  beam_combine<<<1, 32, 0, stream>>>(wsP, wsLP, wsC, (float*)d_out);
}